// NeuromodRNN_14723147890710
// MI455X (gfx1250) — compile-verified
//
#include <hip/hip_runtime.h>
#include <math.h>

typedef float v2f __attribute__((ext_vector_type(2)));
typedef float v8f __attribute__((ext_vector_type(8)));

// Problem sizes (fixed by the reference)
#define BATCH 64
#define TLEN  512
#define IDIM  128
#define HDIM  512
#define ZDIM  32
#define BT    16          // batch tile per workgroup
#define NTILES 32         // HDIM/16
#define RSTR  516         // padded LDS row stride for r arrays (bank-conflict-free)
#define XSTR  132         // padded LDS row stride for x

#define PACKREC (HDIM * HDIM)              // 262144 floats: w_rec fragments
#define PACKIN  (HDIM * IDIM)              // 65536 floats: w_in fragments
// LDS: r | r*V0 | r*V1 | x ping-pong | z ping-pong | fz | s
#define LDS_FLOATS (3*BT*RSTR + 2*BT*XSTR + 2*512 + 512 + 32)

static __device__ __forceinline__ v8f wmma4(v2f a, v2f b, v8f c) {
    // D = A(16x4 f32) * B(4x16 f32) + C(16x16 f32)
    return __builtin_amdgcn_wmma_f32_16x16x4_f32(false, a, false, b, (short)0, c, false, false);
}

// ---------------------------------------------------------------------------
// Prep kernel: repack w_rec + w_in into the per-lane WMMA B-fragment layout:
//   frag[tile][chunk][lane] = {B[k_off][n], B[k_off+1][n]}
// with n = tile*16 + lane%16, k_off = chunk*4 + 2*(lane/16), B[k][n] = W[n][k].
// ---------------------------------------------------------------------------
__global__ void pack_weights(const float* __restrict__ w_rec,  // (H,H)
                             const float* __restrict__ w_in,   // (H,I)
                             float* __restrict__ ws) {
    int tid = blockIdx.x * blockDim.x + threadIdx.x;
    const int recEnt = NTILES * (HDIM/4) * 32;       // 131072 float2 entries
    if (tid < recEnt) {
        int lane  = tid & 31;
        int chunk = (tid >> 5) & 127;
        int tile  = tid >> 12;
        int g = tile * 16 + (lane & 15);
        int h = chunk * 4 + ((lane >> 4) << 1);
        v2f val;
        val.x = w_rec[g * HDIM + h];
        val.y = w_rec[g * HDIM + h + 1];
        *(v2f*)(ws + (size_t)tid * 2) = val;
    } else {
        int id = tid - recEnt;
        if (id < NTILES * (IDIM/4) * 32) {           // 32768 float2 entries
            int lane  = id & 31;
            int chunk = (id >> 5) & 31;
            int tile  = id >> 10;
            int g = tile * 16 + (lane & 15);
            int i = chunk * 4 + ((lane >> 4) << 1);
            v2f val;
            val.x = w_in[g * IDIM + i];
            val.y = w_in[g * IDIM + i + 1];
            *(v2f*)(ws + (size_t)PACKREC + (size_t)id * 2) = val;
        }
    }
}

// ---------------------------------------------------------------------------
// Main persistent recurrence kernel: 4 blocks x 512 threads (16 waves).
// Each block owns 16 batch rows; each wave owns 2 N-tiles of H and keeps the
// membrane state v resident in WMMA D-fragment registers across all T steps.
// A-operands r, r*V0, r*V1 live in LDS; only W/w_in fragments stream from L2.
// ---------------------------------------------------------------------------
__global__ void __launch_bounds__(512) neuromod_rnn(
    const float* __restrict__ inputs,   // (B,T,I)
    const float* __restrict__ w_rec_b,  // (H)
    const float* __restrict__ w_out,    // (H)
    const float* __restrict__ w_outb,   // (1)
    const float* __restrict__ Wzz,      // (Z,Z)
    const float* __restrict__ Wzzb,     // (Z)
    const float* __restrict__ Bz,       // (Z,I)
    const float* __restrict__ Mw,       // (2,Z)
    const float* __restrict__ cb,       // (2)
    const float* __restrict__ U,        // (H,2)
    const float* __restrict__ V,        // (H,2)
    const float* __restrict__ packed,   // d_ws: packed fragments
    float* __restrict__ out)            // [outputs B*T | vT B*H | zT B*Z]
{
    extern __shared__ float lds[];
    float* r_l   = lds;                     // BT * RSTR
    float* rv0_l = r_l   + BT * RSTR;       // BT * RSTR
    float* rv1_l = rv0_l + BT * RSTR;       // BT * RSTR
    float* x_l   = rv1_l + BT * RSTR;       // 2 * BT * XSTR (ping-pong)
    float* z_l   = x_l   + 2 * BT * XSTR;   // 2 * 512 (ping-pong (16,32))
    float* fz_l  = z_l + 1024;              // 512
    float* s_l   = fz_l + 512;              // 32  (stores s - 0.5, layout (16,2))

    const int tid  = threadIdx.x;
    const int wave = tid >> 5;
    const int lane = tid & 31;
    const int nl   = lane & 15;          // A-matrix M index / B-matrix N index
    const int hi   = lane >> 4;
    const int koff = hi << 1;            // K sub-offset {0,2}
    const int b0   = blockIdx.x * BT;

    // zero initial r arrays (= tanh(v0)=0 times anything) and z
    for (int i = tid; i < 3 * BT * RSTR; i += 512) r_l[i] = 0.0f;
    for (int i = tid; i < 1024; i += 512)          z_l[i] = 0.0f;

    // per-wave N-tile constants
    const int tile0 = wave * 2, tile1 = wave * 2 + 1;
    const int n0 = tile0 * 16 + nl, n1 = tile1 * 16 + nl;
    const float U00 = U[n0 * 2 + 0], U01 = U[n0 * 2 + 1];
    const float U10 = U[n1 * 2 + 0], U11 = U[n1 * 2 + 1];
    const float V00 = V[n0 * 2 + 0], V01 = V[n0 * 2 + 1];
    const float V10 = V[n1 * 2 + 0], V11 = V[n1 * 2 + 1];
    const float bn0 = w_rec_b[n0],   bn1 = w_rec_b[n1];
    float wout_r[16];
#pragma unroll
    for (int i = 0; i < 16; ++i) wout_r[i] = w_out[lane + 32 * i];
    const float woutb = w_outb[0];

    // packed B-fragment pointers (per lane)
    const float* pW0 = packed + ((size_t)tile0 * 128) * 64 + lane * 2;
    const float* pW1 = packed + ((size_t)tile1 * 128) * 64 + lane * 2;
    const float* pI0 = packed + PACKREC + ((size_t)tile0 * 32) * 64 + lane * 2;
    const float* pI1 = packed + PACKREC + ((size_t)tile1 * 32) * 64 + lane * 2;
    const float* arp  = r_l   + nl * RSTR + koff;   // A-fragment bases
    const float* avp0 = rv0_l + nl * RSTR + koff;
    const float* avp1 = rv1_l + nl * RSTR + koff;

    // persistent membrane state v, in D-fragment layout
    float vr0[8] = {0,0,0,0,0,0,0,0};
    float vr1[8] = {0,0,0,0,0,0,0,0};

    // prologue: stage x_0 into buffer 0
    const int xb  = (tid * 4) >> 7;          // batch row for b128 staging
    const int xi  = (tid * 4) & 127;         // column
    {
        const float4 x0 = *(const float4*)(inputs + ((size_t)(b0 + xb) * TLEN + 0) * IDIM + xi);
        *(float4*)(x_l + xb * XSTR + xi) = x0;
    }
    __syncthreads();

    int zp = 0, xp = 0;
    for (int t = 0; t < TLEN; ++t) {
        const float* xcur = x_l + xp * (BT * XSTR);
        float*       xnxt = x_l + (1 - xp) * (BT * XSTR);

        // issue next timestep's x load early; completes under the GEMM
        float4 xreg;
        const bool havex = (t + 1 < TLEN);
        if (havex)
            xreg = *(const float4*)(inputs + ((size_t)(b0 + xb) * TLEN + (t + 1)) * IDIM + xi);

        // ---- GEMM: base / Wvr0 / Wvr1 over K=H, + x@w_in^T into base ----
        v8f a00 = {0,0,0,0,0,0,0,0}, a01 = a00, a02 = a00;
        v8f a10 = a00, a11 = a00, a12 = a00;
#pragma unroll 4
        for (int c = 0; c < 128; ++c) {
            v2f ar = *(const v2f*)(arp  + c * 4);
            v2f a0 = *(const v2f*)(avp0 + c * 4);
            v2f a1 = *(const v2f*)(avp1 + c * 4);
            v2f w0 = *(const v2f*)(pW0 + c * 64);
            v2f w1 = *(const v2f*)(pW1 + c * 64);
            a00 = wmma4(ar, w0, a00);
            a10 = wmma4(ar, w1, a10);
            a01 = wmma4(a0, w0, a01);
            a11 = wmma4(a0, w1, a11);
            a02 = wmma4(a1, w0, a02);
            a12 = wmma4(a1, w1, a12);
        }
        {
            const float* axp = xcur + nl * XSTR + koff;
#pragma unroll 4
            for (int c = 0; c < 32; ++c) {
                v2f a  = *(const v2f*)(axp + c * 4);
                v2f i0 = *(const v2f*)(pI0 + c * 64);
                v2f i1 = *(const v2f*)(pI1 + c * 64);
                a00 = wmma4(a, i0, a00);
                a10 = wmma4(a, i1, a10);
            }
        }

        // stage x_{t+1} into the alternate buffer (load already in flight)
        if (havex)
            *(float4*)(xnxt + xb * XSTR + xi) = xreg;

        // ---- neuromodulator z update (tiny, same threads) ---------------
        float* zcur = z_l + zp * 512;
        float* znxt = z_l + (1 - zp) * 512;
        fz_l[tid] = tanhf(zcur[tid]);
        __syncthreads();                               // B2: fz ready
        {
            int bb = tid >> 5, zj = tid & 31;
            float acc = Wzzb[zj];
            const float* wz = Wzz + zj * ZDIM;
            for (int zi = 0; zi < ZDIM; ++zi) acc += fz_l[bb * 32 + zi] * wz[zi];
            const float* bz = Bz + zj * IDIM;
            const float* xr = xcur + bb * XSTR;
            for (int ii = 0; ii < IDIM; ++ii) acc += xr[ii] * bz[ii];
            znxt[tid] = 0.8f * zcur[tid] + 0.2f * acc;
        }
        __syncthreads();                               // B3: z_new ready
        if (tid < 32) {                                // s = sigmoid(z M^T + c) - 0.5
            int k = tid >> 4, bb = tid & 15;
            float acc = cb[k];
            const float* mr = Mw + k * ZDIM;
            for (int zi = 0; zi < ZDIM; ++zi) acc += znxt[bb * 32 + zi] * mr[zi];
            s_l[bb * 2 + k] = 1.0f / (1.0f + expf(-acc)) - 0.5f;
        }
        __syncthreads();                               // B4: s ready, all GEMM r-reads done

        // ---- combine: v update + r_next (and V-scaled copies) into LDS --
#pragma unroll
        for (int j = 0; j < 8; ++j) {
            int m = j + 8 * hi;
            float sm0 = s_l[m * 2 + 0], sm1 = s_l[m * 2 + 1];
            float nv0 = 0.8f * vr0[j] + 0.2f * (a00[j] + bn0 + sm0 * U00 * a01[j] + sm1 * U01 * a02[j]);
            float nv1 = 0.8f * vr1[j] + 0.2f * (a10[j] + bn1 + sm0 * U10 * a11[j] + sm1 * U11 * a12[j]);
            vr0[j] = nv0; vr1[j] = nv1;
            float r0 = tanhf(nv0), r1 = tanhf(nv1);
            int o0 = m * RSTR + tile0 * 16 + nl;
            int o1 = m * RSTR + tile1 * 16 + nl;
            r_l[o0]   = r0;        r_l[o1]   = r1;
            rv0_l[o0] = r0 * V00;  rv0_l[o1] = r1 * V10;
            rv1_l[o0] = r0 * V01;  rv1_l[o1] = r1 * V11;
        }
        __syncthreads();                               // B5: r_next ready

        // ---- readout: y[b,t] = tanh(v)@w_out^T + b  (wave w -> batch w) --
        {
            float p = 0.0f;
            const float* rr = r_l + wave * RSTR;
#pragma unroll
            for (int i = 0; i < 16; ++i) p += rr[lane + 32 * i] * wout_r[i];
#pragma unroll
            for (int off = 16; off > 0; off >>= 1) p += __shfl_down(p, off, 32);
            if (lane == 0) out[(size_t)(b0 + wave) * TLEN + t] = p + woutb;
        }
        __syncthreads();                               // B1': next GEMM may read r arrays
        zp ^= 1; xp ^= 1;
    }

    // ---- final states: vT (B,H) then zT (B,Z), appended after outputs ----
    const size_t OFF_V = (size_t)BATCH * TLEN;
    const size_t OFF_Z = OFF_V + (size_t)BATCH * HDIM;
#pragma unroll
    for (int j = 0; j < 8; ++j) {
        int m = j + 8 * hi;
        out[OFF_V + (size_t)(b0 + m) * HDIM + n0] = vr0[j];
        out[OFF_V + (size_t)(b0 + m) * HDIM + n1] = vr1[j];
    }
    out[OFF_Z + (size_t)(b0 + (tid >> 5)) * ZDIM + (tid & 31)] = z_l[zp * 512 + tid];
}

extern "C" void kernel_launch(void* const* d_in, const int* in_sizes, int n_in,
                              void* d_out, int out_size, void* d_ws, size_t ws_size,
                              hipStream_t stream) {
    (void)in_sizes; (void)n_in; (void)out_size; (void)ws_size;
    const float* inputs = (const float*)d_in[0];
    const float* w_in   = (const float*)d_in[1];
    const float* w_rec  = (const float*)d_in[2];
    const float* w_recb = (const float*)d_in[3];
    const float* w_out  = (const float*)d_in[4];
    const float* w_outb = (const float*)d_in[5];
    const float* Wzz    = (const float*)d_in[6];
    const float* Wzzb   = (const float*)d_in[7];
    const float* Bz     = (const float*)d_in[8];
    const float* Mw     = (const float*)d_in[9];
    const float* cb     = (const float*)d_in[10];
    const float* U      = (const float*)d_in[11];
    const float* V      = (const float*)d_in[12];
    float* ws  = (float*)d_ws;
    float* out = (float*)d_out;

    // 131072 + 32768 float2 entries to pack -> 640 blocks * 256 threads
    pack_weights<<<640, 256, 0, stream>>>(w_rec, w_in, ws);

    size_t ldsBytes = (size_t)LDS_FLOATS * sizeof(float);   // ~122 KB (CDNA5: 320 KB/WGP)
    neuromod_rnn<<<dim3(BATCH / BT), dim3(512), ldsBytes, stream>>>(
        inputs, w_recb, w_out, w_outb, Wzz, Wzzb, Bz, Mw, cb, U, V, ws, out);
}